// GlobalEquivariantLayer_5368709120839
// MI455X (gfx1250) — compile-verified
//
#include <hip/hip_runtime.h>

typedef float v2f __attribute__((ext_vector_type(2)));
typedef float v8f __attribute__((ext_vector_type(8)));

#define NP 1024
#define MP 1024
#define HD 128

__device__ __forceinline__ float fast_silu(float x) {
    // x * sigmoid(x) = x / (1 + exp(-x)); v_exp_f32 + v_rcp_f32
    float e = __expf(-x);
    return x * __builtin_amdgcn_rcpf(1.0f + e);
}

// ---------------------------------------------------------------------------
// dist[n,m] = || lig_coords[n] - poc_coords[m] ||
// ---------------------------------------------------------------------------
__global__ void dist_kernel(const float* __restrict__ lig_coords,
                            const float* __restrict__ poc_coords,
                            float* __restrict__ dist) {
    int idx = blockIdx.x * blockDim.x + threadIdx.x;
    if (idx >= NP * MP) return;
    int n = idx >> 10;          // / MP
    int m = idx & (MP - 1);
    float dx = lig_coords[n * 3 + 0] - poc_coords[m * 3 + 0];
    float dy = lig_coords[n * 3 + 1] - poc_coords[m * 3 + 1];
    float dz = lig_coords[n * 3 + 2] - poc_coords[m * 3 + 2];
    dist[idx] = sqrtf(dx * dx + dy * dy + dz * dz);
}

// ---------------------------------------------------------------------------
// meanpoc[d] = mean over m of poc_coords[m,d]
// ---------------------------------------------------------------------------
__global__ void meanpoc_kernel(const float* __restrict__ poc_coords,
                               float* __restrict__ meanpoc) {
    __shared__ float sb[128 * 3];
    int t = threadIdx.x;  // 128 threads
    float sx = 0.f, sy = 0.f, sz = 0.f;
    for (int m = t; m < MP; m += 128) {
        sx += poc_coords[m * 3 + 0];
        sy += poc_coords[m * 3 + 1];
        sz += poc_coords[m * 3 + 2];
    }
    sb[t * 3 + 0] = sx; sb[t * 3 + 1] = sy; sb[t * 3 + 2] = sz;
    __syncthreads();
    if (t < 3) {
        float s = 0.f;
        for (int i = 0; i < 128; ++i) s += sb[i * 3 + t];
        meanpoc[t] = s * (1.0f / (float)MP);
    }
}

// ---------------------------------------------------------------------------
// WMMA fp32 GEMM: C[M,N] = act(A[M,K] @ W[K,N] + bias[N])
// One wave per 16x16 tile, V_WMMA_F32_16X16X4_F32 over K.
// A 16x4 layout: lane m=lane%16; VGPR v holds K = kk + 2*(lane/16) + v
// B 4x16 layout: lane n=lane%16; VGPR v holds K = kk + 2*(lane/16) + v
// C/D:           lane n=lane%16; VGPR v is row M = v + 8*(lane/16)
// ---------------------------------------------------------------------------
__global__ void wmma_gemm_f32(const float* __restrict__ A,
                              const float* __restrict__ W,
                              const float* __restrict__ bias,
                              float* __restrict__ C,
                              int M, int N, int K, int act) {
    int gwave  = (blockIdx.x * blockDim.x + threadIdx.x) >> 5;
    int lane   = threadIdx.x & 31;
    int tilesN = N >> 4;
    int tm = gwave / tilesN;
    int tn = gwave - tm * tilesN;
    if (tm >= (M >> 4)) return;
    int row0 = tm << 4;
    int col0 = tn << 4;
    int lrow  = lane & 15;          // M (for A) / N (for B,C)
    int khalf = (lane >> 4) << 1;   // 0 or 2

    v8f acc = {};
    for (int kk = 0; kk < K; kk += 4) {
        v2f a, b;
        const float* ap = A + (size_t)(row0 + lrow) * K + kk + khalf;
        a.x = ap[0];
        a.y = ap[1];
        const float* wp = W + (size_t)(kk + khalf) * N + col0 + lrow;
        b.x = wp[0];
        b.y = wp[N];
        acc = __builtin_amdgcn_wmma_f32_16x16x4_f32(
            false, a, false, b, (short)0, acc, false, false);
    }

    int orow = row0 + ((lane >> 4) << 3);
    int ocol = col0 + lrow;
    float bv = bias ? bias[ocol] : 0.0f;
#pragma unroll
    for (int v = 0; v < 8; ++v) {
        float x = acc[v] + bv;
        if (act) x = fast_silu(x);
        C[(size_t)(orow + v) * N + ocol] = x;
    }
}

// ---------------------------------------------------------------------------
// Heavy O(N*M*H) kernel:
// hmean[n,k] = (1/M) * sum_m silu( Apre[n,k] + Bpre[m,k] + dist[n,m]*wd[k] )
// Block = 128 threads (channel k), 8 n-rows per block; dist tile staged in LDS.
// ---------------------------------------------------------------------------
__global__ void silu_acc_kernel(const float* __restrict__ Apre,
                                const float* __restrict__ Bpre,
                                const float* __restrict__ dist,
                                const float* __restrict__ wd_vec,
                                float* __restrict__ hmean) {
    int k  = threadIdx.x;        // 0..127
    int n0 = blockIdx.x << 3;    // 8 rows per block
    float wd = wd_vec[k];
    float a[8], acc[8];
#pragma unroll
    for (int r = 0; r < 8; ++r) {
        a[r]   = Apre[(size_t)(n0 + r) * HD + k];
        acc[r] = 0.0f;
    }
    __shared__ float sD[8 * 128];
    for (int mb = 0; mb < MP; mb += 128) {
        __syncthreads();
#pragma unroll
        for (int r = 0; r < 8; ++r)
            sD[r * 128 + k] = dist[(size_t)(n0 + r) * MP + mb + k];
        __syncthreads();
        for (int mi = 0; mi < 128; ++mi) {
            float b = Bpre[(size_t)(mb + mi) * HD + k];
#pragma unroll
            for (int r = 0; r < 8; ++r) {
                float x = fmaf(sD[r * 128 + mi], wd, a[r] + b);
                acc[r] += fast_silu(x);
            }
        }
    }
    float inv = 1.0f / (float)MP;
#pragma unroll
    for (int r = 0; r < 8; ++r)
        hmean[(size_t)(n0 + r) * HD + k] = acc[r] * inv;
}

// ---------------------------------------------------------------------------
// bias2[j] = bf1[j] + sum_k t_emb[k] * Wf1[128+k, j]   (t_emb row folded in)
// ---------------------------------------------------------------------------
__global__ void bias2_kernel(const float* __restrict__ t_emb,
                             const float* __restrict__ Wf1,
                             const float* __restrict__ bf1,
                             float* __restrict__ bias2) {
    int j = threadIdx.x;  // 128 threads
    float s = bf1[j];
    for (int kk = 0; kk < HD; ++kk)
        s += t_emb[kk] * Wf1[(size_t)(HD + kk) * HD + j];
    bias2[j] = s;
}

// ---------------------------------------------------------------------------
// elementwise c = a + b
// ---------------------------------------------------------------------------
__global__ void add_kernel(const float* __restrict__ a,
                           const float* __restrict__ b,
                           float* __restrict__ c, int n) {
    int i = blockIdx.x * blockDim.x + threadIdx.x;
    if (i < n) c[i] = a[i] + b[i];
}

// ---------------------------------------------------------------------------
// scalar[n] = bc2 + c1[n,:]·Wc2 ; out_coords[n] = lc + scalar*(lc - meanpoc)
// ---------------------------------------------------------------------------
__global__ void scalar_coords_kernel(const float* __restrict__ c1,
                                     const float* __restrict__ Wc2,
                                     const float* __restrict__ bc2,
                                     const float* __restrict__ lig_coords,
                                     const float* __restrict__ meanpoc,
                                     float* __restrict__ out_coords) {
    int n = blockIdx.x * blockDim.x + threadIdx.x;
    if (n >= NP) return;
    float s = bc2[0];
    for (int kk = 0; kk < HD; ++kk)
        s += c1[(size_t)n * HD + kk] * Wc2[kk];
#pragma unroll
    for (int d = 0; d < 3; ++d) {
        float lc = lig_coords[n * 3 + d];
        out_coords[n * 3 + d] = lc + s * (lc - meanpoc[d]);
    }
}

// ---------------------------------------------------------------------------
extern "C" void kernel_launch(void* const* d_in, const int* in_sizes, int n_in,
                              void* d_out, int out_size, void* d_ws, size_t ws_size,
                              hipStream_t stream) {
    const float* lig_feat   = (const float*)d_in[0];
    const float* lig_coords = (const float*)d_in[1];
    const float* poc_feat   = (const float*)d_in[2];
    const float* poc_coords = (const float*)d_in[3];
    const float* t_emb      = (const float*)d_in[4];
    const float* Wm1        = (const float*)d_in[5];
    const float* bm1        = (const float*)d_in[6];
    const float* Wm2        = (const float*)d_in[7];
    const float* bm2        = (const float*)d_in[8];
    const float* Wf1        = (const float*)d_in[9];
    const float* bf1        = (const float*)d_in[10];
    const float* Wf2        = (const float*)d_in[11];
    const float* bf2        = (const float*)d_in[12];
    const float* Wc1        = (const float*)d_in[13];
    const float* bc1        = (const float*)d_in[14];
    const float* Wc2        = (const float*)d_in[15];
    const float* bc2        = (const float*)d_in[16];

    float* out_feat   = (float*)d_out;                 // [1024,128]
    float* out_coords = (float*)d_out + NP * HD;       // [1024,3]

    // workspace layout (floats)
    float* ws      = (float*)d_ws;
    float* dist    = ws;                         // 1024*1024
    float* Apre    = dist  + NP * MP;            // 1024*128
    float* Bpre    = Apre  + NP * HD;
    float* hmean   = Bpre  + MP * HD;
    float* agg     = hmean + NP * HD;
    float* ffin    = agg   + NP * HD;
    float* t1      = ffin  + NP * HD;
    float* c1      = t1    + NP * HD;
    float* bias2   = c1    + NP * HD;            // 128
    float* meanpoc = bias2 + HD;                 // 3

    const float* Wm1_top = Wm1;                  // rows [0,128)   : lig_feat part
    const float* Wm1_mid = Wm1 + HD * HD;        // rows [128,256) : poc_feat part
    const float* Wm1_wd  = Wm1 + 2 * HD * HD;    // row 256        : dist weights

    // GEMM launch config: (M/16)*(N/16) waves, 8 waves (256 thr) per block
    int gemm_waves  = (NP / 16) * (HD / 16);     // 512
    int gemm_blocks = (gemm_waves * 32) / 256;   // 64

    // 1. pairwise distances
    dist_kernel<<<(NP * MP) / 256, 256, 0, stream>>>(lig_coords, poc_coords, dist);
    // 2. mean pocket coords
    meanpoc_kernel<<<1, 128, 0, stream>>>(poc_coords, meanpoc);
    // 3. Apre = lig_feat @ Wm1_top + bm1
    wmma_gemm_f32<<<gemm_blocks, 256, 0, stream>>>(lig_feat, Wm1_top, bm1, Apre, NP, HD, HD, 0);
    // 4. Bpre = poc_feat @ Wm1_mid
    wmma_gemm_f32<<<gemm_blocks, 256, 0, stream>>>(poc_feat, Wm1_mid, nullptr, Bpre, MP, HD, HD, 0);
    // 5. hmean[n,k] = mean_m silu(Apre + Bpre + dist*wd)
    silu_acc_kernel<<<NP / 8, 128, 0, stream>>>(Apre, Bpre, dist, Wm1_wd, hmean);
    // 6. agg = hmean @ Wm2 + bm2
    wmma_gemm_f32<<<gemm_blocks, 256, 0, stream>>>(hmean, Wm2, bm2, agg, NP, HD, HD, 0);
    // 7. fold t_emb row of Wf1 into bias
    bias2_kernel<<<1, 128, 0, stream>>>(t_emb, Wf1, bf1, bias2);
    // 8. ffin = lig_feat + agg
    add_kernel<<<(NP * HD) / 256, 256, 0, stream>>>(lig_feat, agg, ffin, NP * HD);
    // 9. t1 = silu(ffin @ Wf1_top + bias2)
    wmma_gemm_f32<<<gemm_blocks, 256, 0, stream>>>(ffin, Wf1, bias2, t1, NP, HD, HD, 1);
    // 10. new_lig_feat = t1 @ Wf2 + bf2
    wmma_gemm_f32<<<gemm_blocks, 256, 0, stream>>>(t1, Wf2, bf2, out_feat, NP, HD, HD, 0);
    // 11. c1 = silu(agg @ Wc1 + bc1)
    wmma_gemm_f32<<<gemm_blocks, 256, 0, stream>>>(agg, Wc1, bc1, c1, NP, HD, HD, 1);
    // 12. scalar + coord update
    scalar_coords_kernel<<<NP / 256, 256, 0, stream>>>(c1, Wc2, bc2, lig_coords, meanpoc, out_coords);
}